// BidirectNNF_53274774339720
// MI455X (gfx1250) — compile-verified
//
#include <hip/hip_runtime.h>

#define C_   512
#define HS_  192
#define WS_  192
#define HR_  192
#define WR_  192
#define NS_  (HS_ * WS_)   // 36864
#define NR_  (HR_ * WR_)   // 36864
#define ALPHA_ 0.8f
#define TAU_   0.05f

typedef __attribute__((ext_vector_type(2))) float v2f;
typedef __attribute__((ext_vector_type(8))) float v8f;

// ---------------------------------------------------------------------------
// init min/max slots (uint-encoded non-negative floats)
__global__ void bds_init_minmax(unsigned* mm) {
    if (threadIdx.x == 0) { mm[0] = 0xFFFFFFFFu; mm[1] = 0u; }
}

// ---------------------------------------------------------------------------
// Direction A (src->ref vote) inverted into a gather: no atomics, and it
// fully initializes the guide accumulator living in d_out.
__global__ void bds_gatherA(const float* __restrict__ ref,
                            const int*   __restrict__ nnf_sr,
                            float*       __restrict__ out) {
    const int t = blockIdx.x * blockDim.x + threadIdx.x;   // src pixel
    const int c = blockIdx.y;                              // channel
    const int ty = t / WS_, tx = t % WS_;
    const float* refc = ref + (size_t)c * NR_;
    float acc = 0.0f;
#pragma unroll
    for (int dy = -1; dy <= 1; ++dy) {
#pragma unroll
        for (int dx = -1; dx <= 1; ++dx) {
            const int sy = ty - dy, sx = tx - dx;
            if (sy >= 0 && sy < HS_ && sx >= 0 && sx < WS_) {
                const int s  = sy * WS_ + sx;
                const int gy = nnf_sr[2 * s]     + dy;
                const int gx = nnf_sr[2 * s + 1] + dx;
                if (gy >= 0 && gy < HR_ && gx >= 0 && gx < WR_)
                    acc += refc[gy * WR_ + gx];
            }
        }
    }
    out[(size_t)c * NS_ + t] = acc * (1.0f / (float)NS_);
}

__global__ void bds_weight_gatherA(const int* __restrict__ nnf_sr,
                                   float*     __restrict__ weight) {
    const int t = blockIdx.x * blockDim.x + threadIdx.x;
    const int ty = t / WS_, tx = t % WS_;
    float acc = 0.0f;
#pragma unroll
    for (int dy = -1; dy <= 1; ++dy) {
#pragma unroll
        for (int dx = -1; dx <= 1; ++dx) {
            const int sy = ty - dy, sx = tx - dx;
            if (sy >= 0 && sy < HS_ && sx >= 0 && sx < WS_) {
                const int s  = sy * WS_ + sx;
                const int gy = nnf_sr[2 * s]     + dy;
                const int gx = nnf_sr[2 * s + 1] + dx;
                if (gy >= 0 && gy < HR_ && gx >= 0 && gx < WR_)
                    acc += 1.0f;
            }
        }
    }
    weight[t] = acc * (1.0f / (float)NS_);
}

// ---------------------------------------------------------------------------
// Direction B (ref->src vote): data-dependent target -> fp32 scatter atomics.
__global__ void bds_scatterB(const float* __restrict__ ref,
                             const int*   __restrict__ nnf_rs,
                             float*       __restrict__ out) {
    const int r = blockIdx.x * blockDim.x + threadIdx.x;   // ref pixel
    const int c = blockIdx.y;
    const int ryg = r / WR_, rxg = r % WR_;
    const int sy = nnf_rs[2 * r], sx = nnf_rs[2 * r + 1];
    const float* refc = ref + (size_t)c * NR_;
    float* outc = out + (size_t)c * NS_;
    const float wr_w = 2.0f / (float)NR_;
#pragma unroll
    for (int dy = -1; dy <= 1; ++dy) {
#pragma unroll
        for (int dx = -1; dx <= 1; ++dx) {
            const int ty = sy + dy, tx = sx + dx;
            const int gy = ryg + dy, gx = rxg + dx;
            if (ty >= 0 && ty < HS_ && tx >= 0 && tx < WS_ &&
                gy >= 0 && gy < HR_ && gx >= 0 && gx < WR_)
                atomicAdd(&outc[ty * WS_ + tx], refc[gy * WR_ + gx] * wr_w);
        }
    }
}

__global__ void bds_weight_scatterB(const int* __restrict__ nnf_rs,
                                    float*     __restrict__ weight) {
    const int r = blockIdx.x * blockDim.x + threadIdx.x;
    const int ryg = r / WR_, rxg = r % WR_;
    const int sy = nnf_rs[2 * r], sx = nnf_rs[2 * r + 1];
    const float wr_w = 2.0f / (float)NR_;
#pragma unroll
    for (int dy = -1; dy <= 1; ++dy) {
#pragma unroll
        for (int dx = -1; dx <= 1; ++dx) {
            const int ty = sy + dy, tx = sx + dx;
            const int gy = ryg + dy, gx = rxg + dx;
            if (ty >= 0 && ty < HS_ && tx >= 0 && tx < WS_ &&
                gy >= 0 && gy < HR_ && gx >= 0 && gx < WR_)
                atomicAdd(&weight[ty * WS_ + tx], wr_w);
        }
    }
}

// ---------------------------------------------------------------------------
// response[p] = sum_c f_a[c,p]^2 computed as diag(F^T F) via exact-fp32
// V_WMMA_F32_16X16X4_F32. One wave32 per 16-pixel tile, K accumulated 4 at a
// time over 512 channels. Per-lane data for the 16x4 A slab equals the
// per-lane data for the 4x16 B slab (B = A^T), so one v2f serves both.
__global__ void bds_response_wmma(const float* __restrict__ fa,
                                  float*       __restrict__ resp,
                                  unsigned*    __restrict__ mm) {
    const int wave = threadIdx.x >> 5;
    const int lane = threadIdx.x & 31;
    const int tile = blockIdx.x * (blockDim.x >> 5) + wave;
    const int half = lane >> 4;          // lanes 0-15 hold K=0,1; 16-31 K=2,3
    const int id   = lane & 15;          // pixel within tile (M for A, N for B)
    const float* base = fa + (size_t)tile * 16 + id;

    v8f acc = {0.f, 0.f, 0.f, 0.f, 0.f, 0.f, 0.f, 0.f};
    for (int k = 0; k < C_; k += 4) {
        const int ch = k + 2 * half;
        v2f a;
        a[0] = base[(size_t)ch * NS_];
        a[1] = base[(size_t)(ch + 1) * NS_];
        acc = __builtin_amdgcn_wmma_f32_16x16x4_f32(
            /*neg_a=*/false, a, /*neg_b=*/false, a,
            /*c_mod=*/(short)0, acc, /*reuse_a=*/false, /*reuse_b=*/false);
    }

    // C/D layout: VGPR i, lanes 0-15 -> (M=i, N=lane); lanes 16-31 -> (M=i+8,
    // N=lane-16). Diagonal: lanes 0-7 at acc[lane]; lanes 24-31 at acc[lane-24].
    int didx = -1, outp = 0;
    if (lane < 8)        { didx = lane;      outp = tile * 16 + lane; }
    else if (lane >= 24) { didx = lane - 24; outp = tile * 16 + (lane - 16); }
    if (didx >= 0) {
        float dv;
        switch (didx) {
            case 0: dv = acc[0]; break;
            case 1: dv = acc[1]; break;
            case 2: dv = acc[2]; break;
            case 3: dv = acc[3]; break;
            case 4: dv = acc[4]; break;
            case 5: dv = acc[5]; break;
            case 6: dv = acc[6]; break;
            default: dv = acc[7]; break;
        }
        resp[outp] = dv;
        const unsigned u = __float_as_uint(dv);  // dv >= 0: uint order == float order
        atomicMin(&mm[0], u);
        atomicMax(&mm[1], u);
    }
}

// ---------------------------------------------------------------------------
// final blend, in-place on d_out (which holds the guide accumulator)
__global__ void bds_blend(const float*    __restrict__ fa,
                          const float*    __restrict__ weight,
                          const float*    __restrict__ resp,
                          const unsigned* __restrict__ mm,
                          float*          __restrict__ out) {
    const int p = blockIdx.x * blockDim.x + threadIdx.x;
    const int c = blockIdx.y;
    const float mn = __uint_as_float(mm[0]);
    const float mx = __uint_as_float(mm[1]);
    const float rn = (resp[p] - mn) / (mx - mn);
    const float w  = (rn > TAU_) ? ALPHA_ : 0.0f;
    float wt = weight[p];
    wt = (wt == 0.0f) ? 1.0f : wt;
    const size_t i = (size_t)c * NS_ + p;
    const float g = out[i] / wt;
    out[i] = fa[i] * w + g * (1.0f - w);
}

// ---------------------------------------------------------------------------
extern "C" void kernel_launch(void* const* d_in, const int* in_sizes, int n_in,
                              void* d_out, int out_size, void* d_ws, size_t ws_size,
                              hipStream_t stream) {
    (void)in_sizes; (void)n_in; (void)out_size; (void)ws_size;

    const float* ref    = (const float*)d_in[0];   // [C, HR, WR]
    const float* fa     = (const float*)d_in[1];   // [1, C, HS, WS]
    const int*   nnf_sr = (const int*)d_in[2];     // [HS, WS, 2]
    const int*   nnf_rs = (const int*)d_in[3];     // [HR, WR, 2]
    float*       out    = (float*)d_out;           // [1, C, HS, WS] (guide accum)

    float*    weight = (float*)d_ws;               // NS floats
    float*    resp   = weight + NS_;               // NS floats
    unsigned* mm     = (unsigned*)(resp + NS_);    // {min, max} as uint bits

    const dim3 blk(256);
    bds_init_minmax<<<1, 64, 0, stream>>>(mm);
    bds_gatherA<<<dim3(NS_ / 256, C_), blk, 0, stream>>>(ref, nnf_sr, out);
    bds_weight_gatherA<<<dim3(NS_ / 256), blk, 0, stream>>>(nnf_sr, weight);
    bds_scatterB<<<dim3(NR_ / 256, C_), blk, 0, stream>>>(ref, nnf_rs, out);
    bds_weight_scatterB<<<dim3(NR_ / 256), blk, 0, stream>>>(nnf_rs, weight);
    bds_response_wmma<<<dim3(NS_ / 16 / 8), blk, 0, stream>>>(fa, resp, mm);
    bds_blend<<<dim3(NS_ / 256, C_), blk, 0, stream>>>(fa, weight, resp, mm, out);
}